// Mean_29094108463827
// MI455X (gfx1250) — compile-verified
//
#include <hip/hip_runtime.h>
#include <stdint.h>

// scatter_mean for MI455X (gfx1250):
//   sums[n, :] += M[e, :] for dest[e]==n   (L2-resident f32 atomics, no-return)
//   counts[n] += 1
//   M_v = sums / max(counts, 1) ; w[e] = 1 / counts[dest[e]]
// M rows streamed via CDNA5 async global->LDS DMA (ASYNCcnt), depth-4 pipeline:
// ~32K waves x up to 3 outstanding 512B loads ≈ 50 MB in flight > HBM BDP
// (23.3 TB/s x ~2us ≈ 47 MB), so the read stream can saturate memory.

// ---------- CDNA5 async-copy primitives ----------

__device__ __forceinline__ unsigned lds_u32(const void* p) {
    // gfx1250 generic pointer to LDS = {SHARED_BASE, offset32}: low 32 bits
    // are the DS/async-visible LDS byte address.
    return (unsigned)(uintptr_t)p;
}

__device__ __forceinline__ void async_copy_b128(const float* gbase,
                                                unsigned gByteOff,
                                                unsigned ldsByteAddr) {
#if defined(__gfx1250__)
    uint64_t base = (uint64_t)(uintptr_t)gbase;
    // GVS mode: 64-bit SGPR base + per-lane 32-bit byte offset, 16B per lane.
    asm volatile("global_load_async_to_lds_b128 %0, %1, %2"
                 :
                 : "v"(ldsByteAddr), "v"(gByteOff), "s"(base)
                 : "memory");
#else
    (void)gbase; (void)gByteOff; (void)ldsByteAddr;
#endif
}

// Wait until at most n async ops outstanding (n is wave-uniform, 0..3).
__device__ __forceinline__ void wait_async_le(int n) {
#if defined(__gfx1250__)
    switch (n) {
    case 0: asm volatile("s_wait_asynccnt 0x0" ::: "memory"); break;
    case 1: asm volatile("s_wait_asynccnt 0x1" ::: "memory"); break;
    case 2: asm volatile("s_wait_asynccnt 0x2" ::: "memory"); break;
    default: asm volatile("s_wait_asynccnt 0x3" ::: "memory"); break;
    }
#else
    (void)n;
#endif
}

// ---------- kernels ----------

__global__ void zero_f32_kernel(float* __restrict__ p, long long n) {
    long long i = (long long)blockIdx.x * blockDim.x + threadIdx.x;
    if (i < n) p[i] = 0.0f;
}

#define PIPE_DEPTH 4

// Fast path: d_feat == 128. One wave per edge row; 4-slot async DMA pipeline.
__global__ void __launch_bounds__(256)
scatter_accum_d128(const float* __restrict__ M,
                   const int*   __restrict__ dest,
                   float*       __restrict__ sums,
                   float*       __restrict__ counts,
                   int n_edges)
{
    __shared__ __align__(16) float tile[8][PIPE_DEPTH][128]; // 8 waves x 4 slots x 512B = 16KB

    const int lane  = threadIdx.x & 31;
    const int wave  = threadIdx.x >> 5;
    const int gwave = blockIdx.x * 8 + wave;
    const int nwav  = gridDim.x * 8;

    const unsigned laneByte = (unsigned)lane * 16u;
    unsigned lds[PIPE_DEPTH];
#pragma unroll
    for (int s = 0; s < PIPE_DEPTH; ++s)
        lds[s] = lds_u32(&tile[wave][s][0]) + laneByte;

    // Prologue: fill up to DEPTH-1 slots.
    int inflight = 0;
#pragma unroll
    for (int k = 0; k < PIPE_DEPTH - 1; ++k) {
        int ek = gwave + k * nwav;
        if (ek < n_edges) {
            async_copy_b128(M, (unsigned)ek * 512u + laneByte, lds[k]);
            ++inflight;
        }
    }

    int e  = gwave;
    int it = 0;
    while (e < n_edges) {
        // Prefetch DEPTH-1 edges ahead into the slot we'll need then.
        int ep = e + (PIPE_DEPTH - 1) * nwav;
        if (ep < n_edges) {
            async_copy_b128(M, (unsigned)ep * 512u + laneByte,
                            lds[(it + PIPE_DEPTH - 1) & (PIPE_DEPTH - 1)]);
            ++inflight;
        }

        // Async loads complete in order: oldest (current slot) done when
        // outstanding <= inflight-1.
        wait_async_le(inflight - 1);

        const int dn = __builtin_amdgcn_readfirstlane(dest[e]);   // wave-uniform
        const float4 v = *(const float4*)&tile[wave][it & (PIPE_DEPTH - 1)][lane * 4];

        float* o = sums + (size_t)dn * 128u + (unsigned)(lane * 4);
        atomicAdd(o + 0, v.x);
        atomicAdd(o + 1, v.y);
        atomicAdd(o + 2, v.z);
        atomicAdd(o + 3, v.w);
        if (lane == 0) atomicAdd(counts + dn, 1.0f);

        --inflight;
        ++it;
        e += nwav;
    }
}

// Generic fallback for d_feat != 128 (not expected for this reference).
__global__ void scatter_accum_generic(const float* __restrict__ M,
                                      const int*   __restrict__ dest,
                                      float*       __restrict__ sums,
                                      float*       __restrict__ counts,
                                      int n_edges, int d_feat)
{
    long long i = (long long)blockIdx.x * blockDim.x + threadIdx.x;
    long long total = (long long)n_edges * d_feat;
    if (i >= total) return;
    int e = (int)(i / d_feat);
    int f = (int)(i - (long long)e * d_feat);
    int d = dest[e];
    atomicAdd(&sums[(size_t)d * d_feat + f], M[i]);
    if (f == 0) atomicAdd(&counts[d], 1.0f);
}

__global__ void normalize_vec4(float* __restrict__ sums,
                               const float* __restrict__ counts,
                               long long n_vec4, int dvec4)
{
    long long i = (long long)blockIdx.x * blockDim.x + threadIdx.x;
    if (i >= n_vec4) return;
    int node = (int)(i / dvec4);
    float inv = 1.0f / fmaxf(counts[node], 1.0f);
    float4* p = (float4*)sums + i;
    float4 v = *p;
    v.x *= inv; v.y *= inv; v.z *= inv; v.w *= inv;
    *p = v;
}

__global__ void normalize_scalar(float* __restrict__ sums,
                                 const float* __restrict__ counts,
                                 long long n, int d_feat)
{
    long long i = (long long)blockIdx.x * blockDim.x + threadIdx.x;
    if (i >= n) return;
    int node = (int)(i / d_feat);
    sums[i] *= 1.0f / fmaxf(counts[node], 1.0f);
}

__global__ void edge_weights(const int* __restrict__ dest,
                             const float* __restrict__ counts,
                             float* __restrict__ w, int n_edges)
{
    int e = blockIdx.x * blockDim.x + threadIdx.x;
    if (e < n_edges) w[e] = 1.0f / counts[dest[e]];   // referenced nodes have count >= 1
}

// ---------- launcher ----------

extern "C" void kernel_launch(void* const* d_in, const int* in_sizes, int n_in,
                              void* d_out, int out_size, void* d_ws, size_t ws_size,
                              hipStream_t stream)
{
    const float* M    = (const float*)d_in[0];
    const int*   dest = (const int*)d_in[1];

    const int n_edges = in_sizes[1];
    const int d_feat  = in_sizes[0] / n_edges;                 // 128
    const int n_nodes = (out_size - n_edges) / d_feat;         // 40000

    float* sums   = (float*)d_out;                             // [n_nodes, d] accumulated in place
    float* w      = sums + (size_t)n_nodes * d_feat;           // [n_edges]
    float* counts = (float*)d_ws;                              // [n_nodes] scratch

    const long long n_sum = (long long)n_nodes * d_feat;

    // Reset accumulators every call (graph replays must be self-consistent).
    {
        const int thr = 256;
        int b0 = (int)((n_sum + thr - 1) / thr);
        zero_f32_kernel<<<b0, thr, 0, stream>>>(sums, n_sum);
        int b1 = (n_nodes + thr - 1) / thr;
        zero_f32_kernel<<<b1, thr, 0, stream>>>(counts, (long long)n_nodes);
    }

    // Scatter-accumulate.
    if (d_feat == 128) {
        int blocks = (n_edges + 7) / 8;        // 8 waves/block, 1 edge per wave per iter
        if (blocks > 4096) blocks = 4096;      // 32K waves x ~20 iters -> deep DMA pipelining
        scatter_accum_d128<<<blocks, 256, 0, stream>>>(M, dest, sums, counts, n_edges);
    } else {
        const int thr = 256;
        long long total = (long long)n_edges * d_feat;
        int blocks = (int)((total + thr - 1) / thr);
        scatter_accum_generic<<<blocks, thr, 0, stream>>>(M, dest, sums, counts, n_edges, d_feat);
    }

    // Normalize sums -> means (L2-resident RMW, ~41 MB traffic).
    if ((d_feat & 3) == 0) {
        const int thr = 256;
        long long nv = n_sum / 4;
        normalize_vec4<<<(int)((nv + thr - 1) / thr), thr, 0, stream>>>(sums, counts, nv, d_feat / 4);
    } else {
        const int thr = 256;
        normalize_scalar<<<(int)((n_sum + thr - 1) / thr), thr, 0, stream>>>(sums, counts, n_sum, d_feat);
    }

    // Per-edge weights.
    edge_weights<<<(n_edges + 255) / 256, 256, 0, stream>>>(dest, counts, w, n_edges);
}